// FixedNet_62749472194875
// MI455X (gfx1250) — compile-verified
//
#include <hip/hip_runtime.h>
#include <limits.h>

typedef int   v4i __attribute__((ext_vector_type(4)));
typedef float v4f __attribute__((ext_vector_type(4)));

#define N_NODES  1000000
#define N_EDGES  32000000
#define N_GRAPHS 1024
#define N_LAYERS 8
#define OUT_ELEMS (N_GRAPHS * (N_LAYERS + 1))   // 1024 * 9 = 9216
#define RD_MAXSPAN 8

__device__ __forceinline__ void gatomic_add(float* p, float v) {
    // atomicrmw fadd, relaxed, agent scope -> expect global_atomic_add_f32 (no return)
    __hip_atomic_fetch_add(p, v, __ATOMIC_RELAXED, __HIP_MEMORY_SCOPE_AGENT);
}

// ---------------------------------------------------------------------------
// edge kernel FIRST in file so its disasm appears in the feedback snippet:
// need to confirm the fp32 scatter-adds lower to hardware global_atomic_add_f32
// (no-return) and not a CAS retry loop.
//
// agg[dst[e]] += h[src[e]] * ew[e], 4 edges per thread. Edge streams are huge
// (384 MB/layer, re-streamed 8x, > 192MB L2) -> non-temporal B128 loads so the
// 4MB h / 4MB agg working set stays L2-resident for the gathers and atomics.
__global__ void edge_kernel(const int*   __restrict__ src,
                            const int*   __restrict__ dst,
                            const float* __restrict__ ew,
                            const float* __restrict__ h,
                            float*       __restrict__ agg) {
    const v4i* __restrict__ src4 = (const v4i*)src;
    const v4i* __restrict__ dst4 = (const v4i*)dst;
    const v4f* __restrict__ ew4  = (const v4f*)ew;
    const int nq = N_EDGES / 4;
    const int stride = gridDim.x * blockDim.x;
    for (int q = blockIdx.x * blockDim.x + threadIdx.x; q < nq; q += stride) {
        v4i s = __builtin_nontemporal_load(src4 + q);
        v4i d = __builtin_nontemporal_load(dst4 + q);
        v4f w = __builtin_nontemporal_load(ew4  + q);
        gatomic_add(&agg[d.x], h[s.x] * w.x);
        gatomic_add(&agg[d.y], h[s.y] * w.y);
        gatomic_add(&agg[d.z], h[s.z] * w.z);
        gatomic_add(&agg[d.w], h[s.w] * w.w);
    }
}

// ---------------------------------------------------------------------------
// init: xs column 0 = x, agg = 0
__global__ void init_kernel(const float* __restrict__ x,
                            float* __restrict__ xs,
                            float* __restrict__ agg) {
    int i = blockIdx.x * blockDim.x + threadIdx.x;
    if (i < N_NODES) {
        xs[i]  = x[i];
        agg[i] = 0.0f;
    }
}

__global__ void zero_kernel(float* __restrict__ p, int n) {
    int i = blockIdx.x * blockDim.x + threadIdx.x;
    if (i < n) p[i] = 0.0f;
}

// ---------------------------------------------------------------------------
// node kernel: h' = relu(agg*Wn + bn + h*Ws); write next xs column; re-zero
// agg so the next layer's edge kernel starts clean (kernel boundary = sync).
__global__ void node_kernel(float* __restrict__ xs,
                            float* __restrict__ agg,
                            const float* __restrict__ Wn,
                            const float* __restrict__ bn,
                            const float* __restrict__ Ws,
                            int layer) {
    const float wn = Wn[layer];
    const float b  = bn[layer];
    const float ws = Ws[layer];
    int i = blockIdx.x * blockDim.x + threadIdx.x;
    if (i < N_NODES) {
        float a  = agg[i];
        float hh = xs[layer * N_NODES + i];
        float o  = fmaf(a, wn, b) + hh * ws;
        o = o > 0.0f ? o : 0.0f;
        xs[(layer + 1) * N_NODES + i] = o;
        agg[i] = 0.0f;
    }
}

// ---------------------------------------------------------------------------
// readout: out[g][c] = sum over nodes in graph g of xs[c][node].
// graph_ids is sorted and graphs average ~977 nodes, so a 256-node block
// spans <= 2 graphs almost surely: accumulate into LDS (ds_add_f32), then
// flush span*9 global atomics per block instead of 256*9.
__global__ void readout_kernel(const float* __restrict__ xs,
                               const int*   __restrict__ gids,
                               float*       __restrict__ out) {
    __shared__ int   smin, smax;
    __shared__ float sacc[RD_MAXSPAN * 9];

    int i = blockIdx.x * blockDim.x + threadIdx.x;
    bool valid = (i < N_NODES);
    int gid = valid ? gids[i] : 0;

    if (threadIdx.x == 0) { smin = INT_MAX; smax = -1; }
    __syncthreads();
    if (valid) {
        atomicMin(&smin, gid);
        atomicMax(&smax, gid);
    }
    __syncthreads();
    int gmin = smin, gmax = smax;
    if (gmax < 0) return;                       // block-uniform: no valid nodes
    int span = gmax - gmin + 1;

    if (span <= RD_MAXSPAN) {
        for (int j = threadIdx.x; j < span * 9; j += blockDim.x) sacc[j] = 0.0f;
        __syncthreads();
        if (valid) {
            int base = (gid - gmin) * 9;
            #pragma unroll
            for (int c = 0; c < 9; ++c)
                atomicAdd(&sacc[base + c], xs[c * N_NODES + i]);   // ds_add_f32
        }
        __syncthreads();
        for (int j = threadIdx.x; j < span * 9; j += blockDim.x)
            gatomic_add(&out[gmin * 9 + j], sacc[j]);
    } else {
        if (valid) {
            #pragma unroll
            for (int c = 0; c < 9; ++c)
                gatomic_add(&out[gid * 9 + c], xs[c * N_NODES + i]);
        }
    }
}

// ---------------------------------------------------------------------------
extern "C" void kernel_launch(void* const* d_in, const int* in_sizes, int n_in,
                              void* d_out, int out_size, void* d_ws, size_t ws_size,
                              hipStream_t stream) {
    const float* x    = (const float*)d_in[0];
    const int*   src  = (const int*)  d_in[1];
    const int*   dst  = (const int*)  d_in[2];
    const float* ew   = (const float*)d_in[3];
    const int*   gids = (const int*)  d_in[4];
    const float* Wn   = (const float*)d_in[5];
    const float* bn   = (const float*)d_in[6];
    const float* Ws   = (const float*)d_in[7];
    float* out = (float*)d_out;

    // workspace: agg [N], then xs [9][N] column-major (40 MB total)
    float* agg = (float*)d_ws;
    float* xs  = agg + N_NODES;

    const int nodeBlocks = (N_NODES + 255) / 256;

    init_kernel<<<nodeBlocks, 256, 0, stream>>>(x, xs, agg);
    zero_kernel<<<(OUT_ELEMS + 255) / 256, 256, 0, stream>>>(out, OUT_ELEMS);

    for (int l = 0; l < N_LAYERS; ++l) {
        edge_kernel<<<8192, 256, 0, stream>>>(src, dst, ew, xs + (size_t)l * N_NODES, agg);
        node_kernel<<<nodeBlocks, 256, 0, stream>>>(xs, agg, Wn, bn, Ws, l);
    }

    readout_kernel<<<nodeBlocks, 256, 0, stream>>>(xs, gids, out);
}